// Encoder_78597901517411
// MI455X (gfx1250) — compile-verified
//
#include <hip/hip_runtime.h>
#include <hip/hip_bf16.h>
#include <math.h>

// Problem dims (match reference)
#define B_   64
#define T_   512
#define D_   512
#define H_   1024
#define G4_  4096   // 4*H

typedef __attribute__((ext_vector_type(2))) float v2f;
typedef __attribute__((ext_vector_type(8))) float v8f;

__device__ __forceinline__ v8f wmma_f32(v2f a, v2f b, v8f c) {
  // V_WMMA_F32_16X16X4_F32 : D = A(16x4) * B(4x16) + C(16x16), all f32
  return __builtin_amdgcn_wmma_f32_16x16x4_f32(false, a, false, b, (short)0, c, false, false);
}

__device__ __forceinline__ float sigmoidf_(float x) {
  return 1.0f / (1.0f + __expf(-x));
}

// ---------------------------------------------------------------------------
// Kernel 0: zero the grid-barrier state (d_ws is poisoned by the harness).
// ---------------------------------------------------------------------------
__global__ void init_sync_kernel(unsigned* s) {
  if (threadIdx.x < 2) s[threadIdx.x] = 0u;
}

// ---------------------------------------------------------------------------
// Kernel 1: XW = E[ids] @ W + b   (M=32768, K=512, N=4096), f32 WMMA.
// 128x128 tile / workgroup, 256 threads (8 wave32), K-chunk = 32 in LDS.
// B tile is stored K-pair-interleaved: Bl[k2][col][2] holds
// (W[2*k2][col], W[2*k2+1][col]) contiguously -> each B fragment is one
// aligned ds_load_b64 directly into an even VGPR pair (no v_mov shuffles).
// Column pad 144: 2*144 = 288 = 32 (mod 64 banks) -> the two half-waves
// (k2 vs k2+1) hit disjoint 32-bank windows.
// ---------------------------------------------------------------------------
#define K1_BM   128
#define K1_BN   128
#define K1_BK   32
#define K1_LDA  36     // padded A stride (floats)
#define K1_LDBP 144    // padded B column count (pair-interleaved)

__global__ __launch_bounds__(256) void xw_gemm_kernel(
    const int* __restrict__ ids, const float* __restrict__ E,
    const float* __restrict__ W, const float* __restrict__ bias,
    float* __restrict__ XW)
{
  __shared__ float Al[K1_BM * K1_LDA];              // 18.0 KB
  __shared__ float Bl[(K1_BK / 2) * K1_LDBP * 2];   // 18.0 KB

  const int tid  = threadIdx.x;
  const int lane = tid & 31;
  const int wave = tid >> 5;
  const int m0   = blockIdx.y * K1_BM;
  const int n0   = blockIdx.x * K1_BN;

  v8f acc[8];
#pragma unroll
  for (int c = 0; c < 8; ++c)
#pragma unroll
    for (int i = 0; i < 8; ++i) acc[c][i] = 0.0f;

  for (int kb = 0; kb < D_; kb += K1_BK) {
    // ---- A tile: embedding gather, 128 rows x 32 cols
    {
      const int rbase = tid >> 3;        // 0..31
      const int c4    = (tid & 7) * 4;   // 0,4,...,28
#pragma unroll
      for (int p = 0; p < 4; ++p) {
        const int r  = p * 32 + rbase;
        const int id = ids[m0 + r];
        const float4 v = *(const float4*)(E + (size_t)id * D_ + kb + c4);
        *(float4*)&Al[r * K1_LDA + c4] = v;
      }
    }
    // ---- B tile: W[kb..kb+32)[n0..n0+128), pair-interleaved
    {
#pragma unroll
      for (int p = 0; p < 4; ++p) {
        const int idx = p * 256 + tid;   // 16 k2-rows x 64 col-pairs
        const int k2  = idx >> 6;        // 0..15
        const int c   = (idx & 63) * 2;  // even col 0..126
        const float2 u0 = *(const float2*)(W + (size_t)(kb + 2 * k2)     * G4_ + n0 + c);
        const float2 u1 = *(const float2*)(W + (size_t)(kb + 2 * k2 + 1) * G4_ + n0 + c);
        float4 v;
        v.x = u0.x; v.y = u1.x; v.z = u0.y; v.w = u1.y;
        *(float4*)&Bl[(k2 * K1_LDBP + c) * 2] = v;
      }
    }
    __syncthreads();

#pragma unroll
    for (int k4 = 0; k4 < K1_BK / 4; ++k4) {
      const int kk   = k4 * 4 + ((lane >> 4) << 1);  // A layout: v0=K0|K2, v1=K1|K3
      const int k2f  = k4 * 2 + (lane >> 4);         // B pair row
      const int arow = wave * 16 + (lane & 15);
      const v2f a = *(const v2f*)&Al[arow * K1_LDA + kk];
#pragma unroll
      for (int c = 0; c < 8; ++c) {
        const int bcol = c * 16 + (lane & 15);
        const v2f b = *(const v2f*)&Bl[(k2f * K1_LDBP + bcol) * 2];
        acc[c] = wmma_f32(a, b, acc[c]);
      }
    }
    __syncthreads();
  }

  // ---- epilogue: + bias, store.  C layout: vgpr r -> M = r + 8*(lane>>4), N = lane&15
  const int rb2 = (lane >> 4) * 8;
  const int col = lane & 15;
#pragma unroll
  for (int c = 0; c < 8; ++c) {
    const int gcol = n0 + c * 16 + col;
    const float bv = bias[gcol];
#pragma unroll
    for (int r = 0; r < 8; ++r) {
      const int grow = m0 + wave * 16 + rb2 + r;
      XW[(size_t)grow * G4_ + gcol] = acc[c][r] + bv;
    }
  }
}

// ---------------------------------------------------------------------------
// Kernel 2: persistent LSTM scan. 64 workgroups, each owns 16 hidden columns
// (=> 64 output columns: {i,f,g,o} x 16). Per step: z = xw_t + h @ U, gates,
// state update, grid barrier. U slice (1024x64) lives transposed in LDS for
// the whole kernel (263 KB, stride 1028 == 4 mod 64 banks); h chunks (64x128)
// staged per step (33 KB). h double-buffered in d_ws; c is WG-private.
// ---------------------------------------------------------------------------
#define NWG2   64
#define K2_CH  128
#define K2_LH  132    // h-chunk LDS stride
#define K2_LU  1028   // U^T LDS stride

__device__ __forceinline__ void grid_barrier(unsigned* cnt, unsigned* gen, unsigned nwg) {
  __threadfence();
  __syncthreads();
  if (threadIdx.x == 0) {
    const unsigned g = __hip_atomic_load(gen, __ATOMIC_ACQUIRE, __HIP_MEMORY_SCOPE_AGENT);
    const unsigned prev =
        __hip_atomic_fetch_add(cnt, 1u, __ATOMIC_ACQ_REL, __HIP_MEMORY_SCOPE_AGENT);
    if (prev == nwg - 1u) {
      __hip_atomic_store(cnt, 0u, __ATOMIC_RELAXED, __HIP_MEMORY_SCOPE_AGENT);
      __hip_atomic_fetch_add(gen, 1u, __ATOMIC_ACQ_REL, __HIP_MEMORY_SCOPE_AGENT);
    } else {
      while (__hip_atomic_load(gen, __ATOMIC_ACQUIRE, __HIP_MEMORY_SCOPE_AGENT) == g) {
        __builtin_amdgcn_s_sleep(1);
      }
    }
  }
  __syncthreads();
}

__global__ __launch_bounds__(256) void lstm_scan_kernel(
    const int* __restrict__ ids, const float* __restrict__ U,
    const float* __restrict__ XW, float* __restrict__ out,
    float* __restrict__ h0, float* __restrict__ h1,
    float* __restrict__ c_ws, unsigned* __restrict__ sync)
{
  extern __shared__ float smem[];
  float* Ut = smem;                  // [64][K2_LU]  U^T slice, persistent
  float* Hl = smem + 64 * K2_LU;     // [64][K2_LH]  h chunk, reused as z-buffer
  float* Zl = Hl;                    // z[4 gates][64 rows][16 cols] = 4096 floats

  const int tid  = threadIdx.x;
  const int lane = tid & 31;
  const int wave = tid >> 5;
  const int n0   = blockIdx.x * 16;        // hidden-column base for this WG
  const int rt   = wave & 3;               // row tile (batch rows rt*16..)
  const int g0   = (wave >> 2) << 1;       // gate pair {g0, g0+1}

  // ---- stage U^T: Ut[g*16+j][k] = U[k][g*H + n0 + j]
  {
    const int ct = tid & 15;               // float4 slot: gate = ct>>2, j4 = (ct&3)*4
    const int g  = ct >> 2;
    const int j4 = (ct & 3) * 4;
    const int kt = tid >> 4;               // 0..15
    for (int p = 0; p < 64; ++p) {
      const int k = p * 16 + kt;
      const float4 v = *(const float4*)(U + (size_t)k * G4_ + g * H_ + n0 + j4);
      const int c = g * 16 + j4;
      Ut[(c + 0) * K2_LU + k] = v.x;
      Ut[(c + 1) * K2_LU + k] = v.y;
      Ut[(c + 2) * K2_LU + k] = v.z;
      Ut[(c + 3) * K2_LU + k] = v.w;
    }
  }
  // ---- zero our columns of h0/h1/c
  {
    const int e = tid * 4;                 // 1024 state elems, 4 per thread
    const int b = e >> 4;
    const int j = e & 15;
    const float4 z4 = make_float4(0.f, 0.f, 0.f, 0.f);
    *(float4*)(h0 + (size_t)b * H_ + n0 + j)   = z4;
    *(float4*)(h1 + (size_t)b * H_ + n0 + j)   = z4;
    *(float4*)(c_ws + (size_t)b * H_ + n0 + j) = z4;
  }
  grid_barrier(sync, sync + 1, NWG2);

#pragma unroll 1
  for (int t = 0; t < T_; ++t) {
    const float* hr = (t & 1) ? h1 : h0;   // read buffer (stable this step)
    float*       hw = (t & 1) ? h0 : h1;   // write buffer

    // ---- prefetch this step's XW gate segments (global_prefetch_b8);
    //      the recurrent GEMM below gives them a full chunk-loop of lead time.
    {
      const int e = tid * 4;
      const int b = e >> 4;
      const int j = e & 15;
      const float* xwrow = XW + ((size_t)b * T_ + t) * G4_;
      __builtin_prefetch(xwrow + 0 * H_ + n0 + j, 0, 0);
      __builtin_prefetch(xwrow + 1 * H_ + n0 + j, 0, 0);
      __builtin_prefetch(xwrow + 2 * H_ + n0 + j, 0, 0);
      __builtin_prefetch(xwrow + 3 * H_ + n0 + j, 0, 0);
    }

    v8f acc0, acc1;
#pragma unroll
    for (int i = 0; i < 8; ++i) { acc0[i] = 0.0f; acc1[i] = 0.0f; }

#pragma unroll 1
    for (int kc = 0; kc < H_; kc += K2_CH) {
      __syncthreads();                     // Hl/Zl region free for reuse
      // stage h chunk: 64 x 128 floats
#pragma unroll
      for (int p = 0; p < 8; ++p) {
        const int idx = p * 256 + tid;
        const int b   = idx >> 5;
        const int c4  = (idx & 31) << 2;
        const float4 v = *(const float4*)(hr + (size_t)b * H_ + kc + c4);
        *(float4*)&Hl[b * K2_LH + c4] = v;
      }
      __syncthreads();

#pragma unroll
      for (int k4 = 0; k4 < K2_CH / 4; ++k4) {
        const int kk   = k4 * 4 + ((lane >> 4) << 1);
        const int arow = rt * 16 + (lane & 15);
        const v2f a = *(const v2f*)&Hl[arow * K2_LH + kk];
        const int gk = kc + kk;
        const int c0 = g0 * 16 + (lane & 15);
        const int c1 = c0 + 16;
        const v2f b0 = *(const v2f*)&Ut[c0 * K2_LU + gk];
        const v2f b1 = *(const v2f*)&Ut[c1 * K2_LU + gk];
        acc0 = wmma_f32(a, b0, acc0);
        acc1 = wmma_f32(a, b1, acc1);
      }
    }
    __syncthreads();
    // ---- park z tiles in LDS (C layout: M = r + 8*(lane>>4), N = lane&15)
    {
      const int col = lane & 15;
      const int rb  = (lane >> 4) * 8;
#pragma unroll
      for (int r = 0; r < 8; ++r) {
        const int brow = rt * 16 + rb + r;
        Zl[((g0 + 0) * 64 + brow) * 16 + col] = acc0[r];
        Zl[((g0 + 1) * 64 + brow) * 16 + col] = acc1[r];
      }
    }
    __syncthreads();
    // ---- gates + state update: 1024 elems, 4 per thread
    {
      const int e = tid * 4;
      const int b = e >> 4;
      const int j = e & 15;
      const int m = (ids[b * T_ + t] != 0);
      const float* xwrow = XW + ((size_t)b * T_ + t) * G4_;
      const float4 xi = *(const float4*)(xwrow + 0 * H_ + n0 + j);
      const float4 xf = *(const float4*)(xwrow + 1 * H_ + n0 + j);
      const float4 xg = *(const float4*)(xwrow + 2 * H_ + n0 + j);
      const float4 xo = *(const float4*)(xwrow + 3 * H_ + n0 + j);
      const float* xiv = (const float*)&xi;
      const float* xfv = (const float*)&xf;
      const float* xgv = (const float*)&xg;
      const float* xov = (const float*)&xo;
#pragma unroll
      for (int q = 0; q < 4; ++q) {
        const int jj = j + q;
        const float iv = sigmoidf_(Zl[(0 * 64 + b) * 16 + jj] + xiv[q]);
        const float fv = sigmoidf_(Zl[(1 * 64 + b) * 16 + jj] + xfv[q]);
        const float gv = sigmoidf_(Zl[(2 * 64 + b) * 16 + jj] + xgv[q]);
        const float ov = sigmoidf_(Zl[(3 * 64 + b) * 16 + jj] + xov[q]);
        const size_t sidx = (size_t)b * H_ + n0 + jj;
        const float cold = c_ws[sidx];
        const float hold = hr[sidx];
        const float cnew = fv * cold + iv * gv;
        const float hnew = ov * sigmoidf_(cnew);
        const float hsel = m ? hnew : hold;
        const float csel = m ? cnew : cold;
        hw[sidx]   = hsel;
        c_ws[sidx] = csel;
        out[((size_t)b * T_ + t) * H_ + n0 + jj] = hsel;          // outs[b][t][:]
        if (t == T_ - 1) {
          out[(size_t)B_ * T_ * H_ + sidx]                   = hsel; // hF
          out[(size_t)B_ * T_ * H_ + (size_t)B_ * H_ + sidx] = csel; // cF
        }
      }
    }
    grid_barrier(sync, sync + 1, NWG2);
  }
}

// ---------------------------------------------------------------------------
// Host launch. d_ws layout (floats): [0..255] barrier state | h0 | h1 | c | XW.
// Needs ~537 MB of scratch (XW = 32768 x 4096 f32 is the LSTM pre-GEMM).
// ---------------------------------------------------------------------------
extern "C" void kernel_launch(void* const* d_in, const int* in_sizes, int n_in,
                              void* d_out, int out_size, void* d_ws, size_t ws_size,
                              hipStream_t stream) {
  const int*   ids  = (const int*)d_in[0];
  const float* E    = (const float*)d_in[1];
  const float* W    = (const float*)d_in[2];
  const float* U    = (const float*)d_in[3];
  const float* bias = (const float*)d_in[4];
  float*       out  = (float*)d_out;

  unsigned* sync = (unsigned*)d_ws;
  float* base = (float*)d_ws;
  float* h0 = base + 256;
  float* h1 = h0 + (size_t)B_ * H_;
  float* cw = h1 + (size_t)B_ * H_;
  float* xw = cw + (size_t)B_ * H_;

  init_sync_kernel<<<1, 64, 0, stream>>>(sync);

  dim3 g1(G4_ / K1_BN, (B_ * T_) / K1_BM);   // (32, 256)
  xw_gemm_kernel<<<g1, 256, 0, stream>>>(ids, E, W, bias, xw);

  const size_t smem2 = (size_t)(64 * K2_LU + 64 * K2_LH) * sizeof(float); // ~290 KB
  lstm_scan_kernel<<<NWG2, 256, smem2, stream>>>(ids, U, xw, out, h0, h1, cw, sync);
}